// FidelityTripletLoss_30133490549324
// MI455X (gfx1250) — compile-verified
//
#include <hip/hip_runtime.h>
#include <hip/hip_bf16.h>

#define NQ 4
#define QDIM 16
#define N_LAYERS 2
#define MARGIN_F 0.3f
#define BATCH_N 524288
#define TPB 256
#define NBLOCKS 512
#define PER_THREAD 4   // 512 * 256 * 4 == 524288, exact cover, no tail divergence

typedef float v2f __attribute__((ext_vector_type(2)));
typedef float v8f __attribute__((ext_vector_type(8)));

// ---------------------------------------------------------------------------
// Wave-level sum via V_WMMA_F32_16X16X4_F32.
// A layout (32-bit 16x4): lanes 0-15 -> {VGPR0:K=0, VGPR1:K=1},
//                         lanes 16-31 -> {VGPR0:K=2, VGPR1:K=3}.
// Put val in VGPR0, 0 in VGPR1; B = all ones (layout-independent).
// D[m,n] = val_m + val_{m+16}.  Lane L sums its 8 D rows, then one xor-16
// shuffle yields the full 32-lane sum in every lane.
// Requires EXEC all ones (call with no divergence).
// ---------------------------------------------------------------------------
__device__ __forceinline__ float wave_reduce_wmma(float val) {
  v2f a; a[0] = val;  a[1] = 0.0f;
  v2f b; b[0] = 1.0f; b[1] = 1.0f;
  v8f c = {};
  c = __builtin_amdgcn_wmma_f32_16x16x4_f32(
      /*neg_a=*/false, a, /*neg_b=*/false, b,
      /*c_mod=*/(short)0, c, /*reuse_a=*/false, /*reuse_b=*/false);
  float s = ((c[0] + c[1]) + (c[2] + c[3])) + ((c[4] + c[5]) + (c[6] + c[7]));
  s += __shfl_xor(s, 16, 32);
  return s;
}

// ---------------------------------------------------------------------------
// Quantum gates on a 16-amplitude complex state held entirely in VGPRs.
// Reference reshape (B, 2^q, 2, 2^(NQ-q-1)) => qubit q lives at bit (3-q).
// ---------------------------------------------------------------------------
template <int Q>
__device__ __forceinline__ void apply_ry(float re[QDIM], float im[QDIM], float half_theta) {
  float s, c;
  __sincosf(half_theta, &s, &c);
  constexpr int T = 8 >> Q;
#pragma unroll
  for (int i = 0; i < QDIM; ++i) {
    if ((i & T) == 0) {
      const int j = i | T;
      const float a0r = re[i], a0i = im[i];
      const float a1r = re[j], a1i = im[j];
      re[i] = c * a0r - s * a1r;
      im[i] = c * a0i - s * a1i;
      re[j] = s * a0r + c * a1r;
      im[j] = s * a0i + c * a1i;
    }
  }
}

template <int Q>
__device__ __forceinline__ void apply_rz(float re[QDIM], float im[QDIM], float half_theta) {
  float s, c;
  __sincosf(half_theta, &s, &c);
  constexpr int T = 8 >> Q;
#pragma unroll
  for (int i = 0; i < QDIM; ++i) {
    const float r = re[i], m = im[i];
    if ((i & T) == 0) {      // a *= (c - i*s)
      re[i] = c * r + s * m;
      im[i] = c * m - s * r;
    } else {                 // a *= (c + i*s)
      re[i] = c * r - s * m;
      im[i] = c * m + s * r;
    }
  }
}

template <int CQ, int TQ>
__device__ __forceinline__ void apply_cnot(float re[QDIM], float im[QDIM]) {
  constexpr int CB = 8 >> CQ;
  constexpr int TB = 8 >> TQ;
#pragma unroll
  for (int i = 0; i < QDIM; ++i) {
    if ((i & CB) != 0 && (i & TB) == 0) {
      const int j = i | TB;
      const float tr = re[i]; re[i] = re[j]; re[j] = tr;
      const float ti = im[i]; im[i] = im[j]; im[j] = ti;
    }
  }
}

__device__ __forceinline__ void load8(const float* __restrict__ p, float out[8]) {
  const float4 v0 = ((const float4*)p)[0];
  const float4 v1 = ((const float4*)p)[1];
  out[0] = v0.x; out[1] = v0.y; out[2] = v0.z; out[3] = v0.w;
  out[4] = v1.x; out[5] = v1.y; out[6] = v1.z; out[7] = v1.w;
}

__device__ __forceinline__ void simulate(const float ry[8], const float rz[8],
                                         float re[QDIM], float im[QDIM]) {
#pragma unroll
  for (int i = 0; i < QDIM; ++i) { re[i] = 0.0f; im[i] = 0.0f; }
  re[0] = 1.0f;
#pragma unroll
  for (int l = 0; l < N_LAYERS; ++l) {
    apply_ry<0>(re, im, 0.5f * ry[l * 4 + 0]); apply_rz<0>(re, im, 0.5f * rz[l * 4 + 0]);
    apply_ry<1>(re, im, 0.5f * ry[l * 4 + 1]); apply_rz<1>(re, im, 0.5f * rz[l * 4 + 1]);
    apply_ry<2>(re, im, 0.5f * ry[l * 4 + 2]); apply_rz<2>(re, im, 0.5f * rz[l * 4 + 2]);
    apply_ry<3>(re, im, 0.5f * ry[l * 4 + 3]); apply_rz<3>(re, im, 0.5f * rz[l * 4 + 3]);
    apply_cnot<0, 1>(re, im);
    apply_cnot<1, 2>(re, im);
    apply_cnot<2, 3>(re, im);
  }
}

__device__ __forceinline__ float fidelity(const float ar[QDIM], const float ai[QDIM],
                                          const float br[QDIM], const float bi[QDIM]) {
  float ovr = 0.0f, ovi = 0.0f;
#pragma unroll
  for (int i = 0; i < QDIM; ++i) {
    ovr += ar[i] * br[i] + ai[i] * bi[i];   // Re(conj(a)*b)
    ovi += ar[i] * bi[i] - ai[i] * br[i];   // Im(conj(a)*b)
  }
  return ovr * ovr + ovi * ovi;
}

// ---------------------------------------------------------------------------
// Pass 1: rolled loop over PER_THREAD batch elements (keeps code size and
// VGPR pressure down -> no spills, fits I$), full circuit sim in registers,
// WMMA wave reduction + LDS block reduction -> one partial per block.
// ---------------------------------------------------------------------------
__global__ __launch_bounds__(TPB) void triplet_loss_pass1(
    const float* __restrict__ a_ry, const float* __restrict__ a_rz,
    const float* __restrict__ p_ry, const float* __restrict__ p_rz,
    const float* __restrict__ n_ry, const float* __restrict__ n_rz,
    float* __restrict__ partials) {
  const int nthreads = NBLOCKS * TPB;
  const int tid = blockIdx.x * TPB + threadIdx.x;

  float acc = 0.0f;
#pragma unroll 1
  for (int k = 0; k < PER_THREAD; ++k) {
    const int b = tid + k * nthreads;
    const int base = b * 8;

    if (k + 1 < PER_THREAD) {  // uniform branch: stream next iteration into cache
      const int nb = (b + nthreads) * 8;
      __builtin_prefetch(a_ry + nb, 0, 0);
      __builtin_prefetch(a_rz + nb, 0, 0);
      __builtin_prefetch(p_ry + nb, 0, 0);
      __builtin_prefetch(p_rz + nb, 0, 0);
      __builtin_prefetch(n_ry + nb, 0, 0);
      __builtin_prefetch(n_rz + nb, 0, 0);
    }

    float ry[8], rz[8];
    float are[QDIM], aim[QDIM], cre[QDIM], cim[QDIM];

    load8(a_ry + base, ry); load8(a_rz + base, rz);
    simulate(ry, rz, are, aim);

    load8(p_ry + base, ry); load8(p_rz + base, rz);
    simulate(ry, rz, cre, cim);
    const float fid_pos = fidelity(are, aim, cre, cim);

    load8(n_ry + base, ry); load8(n_rz + base, rz);
    simulate(ry, rz, cre, cim);
    const float fid_neg = fidelity(are, aim, cre, cim);

    acc += fmaxf(MARGIN_F - fid_pos + fid_neg, 0.0f);
  }

  // No divergence above at the thread level -> EXEC is all ones here.
  const float wsum = wave_reduce_wmma(acc);

  __shared__ float smem[TPB / 32];
  const int lane = threadIdx.x & 31;
  const int wave = threadIdx.x >> 5;
  if (lane == 0) smem[wave] = wsum;
  __syncthreads();
  if (threadIdx.x == 0) {
    float t = 0.0f;
#pragma unroll
    for (int w = 0; w < TPB / 32; ++w) t += smem[w];
    partials[blockIdx.x] = t;
  }
}

// ---------------------------------------------------------------------------
// Pass 2: reduce NBLOCKS partials (multiple of TPB -> uniform loop, EXEC full)
// and write the mean. Deterministic tree, no atomics.
// ---------------------------------------------------------------------------
__global__ __launch_bounds__(TPB) void triplet_loss_pass2(
    const float* __restrict__ partials, float* __restrict__ out) {
  float acc = 0.0f;
#pragma unroll
  for (int i = threadIdx.x; i < NBLOCKS; i += TPB) acc += partials[i];

  const float wsum = wave_reduce_wmma(acc);

  __shared__ float smem[TPB / 32];
  const int lane = threadIdx.x & 31;
  const int wave = threadIdx.x >> 5;
  if (lane == 0) smem[wave] = wsum;
  __syncthreads();
  if (threadIdx.x == 0) {
    float t = 0.0f;
#pragma unroll
    for (int w = 0; w < TPB / 32; ++w) t += smem[w];
    out[0] = t * (1.0f / (float)BATCH_N);
  }
}

extern "C" void kernel_launch(void* const* d_in, const int* in_sizes, int n_in,
                              void* d_out, int out_size, void* d_ws, size_t ws_size,
                              hipStream_t stream) {
  const float* a_ry = (const float*)d_in[0];
  const float* a_rz = (const float*)d_in[1];
  const float* p_ry = (const float*)d_in[2];
  const float* p_rz = (const float*)d_in[3];
  const float* n_ry = (const float*)d_in[4];
  const float* n_rz = (const float*)d_in[5];
  float* out = (float*)d_out;
  float* partials = (float*)d_ws;  // NBLOCKS floats of scratch

  triplet_loss_pass1<<<NBLOCKS, TPB, 0, stream>>>(a_ry, a_rz, p_ry, p_rz, n_ry, n_rz, partials);
  triplet_loss_pass2<<<1, TPB, 0, stream>>>(partials, out);
}